// ConditionalFeedForward_47605417509024
// MI455X (gfx1250) — compile-verified
//
#include <hip/hip_runtime.h>

typedef __attribute__((ext_vector_type(2))) float v2f;
typedef __attribute__((ext_vector_type(8))) float v8f;

#define NUM_EXPERTS 8
#define INTER 7168
#define DIM 2048
#define SEQ 16
#define TOPK 2

// D = A(16x4) * B(4x16) + C, fp32 WMMA.
static __device__ __forceinline__ v8f wmma_f32(v2f a, v2f b, v8f c) {
  return __builtin_amdgcn_wmma_f32_16x16x4_f32(false, a, false, b, (short)0, c,
                                               false, false);
}

// Kernel 1: h[e][t][o] = silu(x . w1[e,o,:]) * (x . w3[e,o,:])
// One wave computes a 16(token) x 16(o) tile for expert e with two accumulators
// (w1 path and w3 path) sharing the A fragment.
__global__ __launch_bounds__(256) void moe_h_kernel(
    const float* __restrict__ x,   // [SEQ, DIM]
    const float* __restrict__ w1,  // [E, INTER, DIM]
    const float* __restrict__ w3,  // [E, INTER, DIM]
    float* __restrict__ h) {       // [E, SEQ, INTER]
  const int e    = blockIdx.y;
  const int wave = threadIdx.x >> 5;
  const int lane = threadIdx.x & 31;
  const int half = lane >> 4;   // selects K pair {0,1} vs {2,3}
  const int lr   = lane & 15;   // row (A: token M, B: output col N)
  const int o    = blockIdx.x * 128 + wave * 16 + lr;

  const float* xr  = x  + (size_t)lr * DIM;                  // token row
  const float* w1r = w1 + ((size_t)e * INTER + o) * DIM;     // weight row o
  const float* w3r = w3 + ((size_t)e * INTER + o) * DIM;
  const int koff = 2 * half;

  v8f acc1 = {};
  v8f acc3 = {};
  for (int k0 = 0; k0 < DIM; k0 += 8) {
    v2f a0 = *(const v2f*)(xr  + k0 + koff);
    v2f p0 = *(const v2f*)(w1r + k0 + koff);
    v2f q0 = *(const v2f*)(w3r + k0 + koff);
    acc1 = wmma_f32(a0, p0, acc1);
    acc3 = wmma_f32(a0, q0, acc3);
    v2f a1 = *(const v2f*)(xr  + k0 + 4 + koff);
    v2f p1 = *(const v2f*)(w1r + k0 + 4 + koff);
    v2f q1 = *(const v2f*)(w3r + k0 + 4 + koff);
    acc1 = wmma_f32(a1, p1, acc1);
    acc3 = wmma_f32(a1, q1, acc3);
  }

  // C/D layout: VGPR j, lanes 0-15 -> M=j; lanes 16-31 -> M=j+8; N = lr.
  float* hb = h + (size_t)e * SEQ * INTER;
#pragma unroll
  for (int j = 0; j < 8; ++j) {
    const int t = j + 8 * half;
    const float s1 = acc1[j];
    const float s3 = acc3[j];
    const float sig = 1.0f / (1.0f + __expf(-s1));
    hb[(size_t)t * INTER + o] = s1 * sig * s3;
  }
}

// Kernel 2: out[t][k][i] = h[e(t,k)][t][:] . w2[e(t,k), i, :]
// Dense per-expert GEMM; each lane scatters its finished C elements directly
// into the routed [T, K, D] output slots whose expert matches.
__global__ __launch_bounds__(256) void moe_out_kernel(
    const float* __restrict__ h,    // [E, SEQ, INTER]
    const float* __restrict__ w2,   // [E, DIM, INTER]
    const int* __restrict__ idx,    // [SEQ, TOPK]
    float* __restrict__ out) {      // [SEQ, TOPK, DIM]
  const int e    = blockIdx.y;
  const int wave = threadIdx.x >> 5;
  const int lane = threadIdx.x & 31;
  const int half = lane >> 4;
  const int lr   = lane & 15;
  const int i    = blockIdx.x * 128 + wave * 16 + lr;

  const float* hr  = h  + ((size_t)e * SEQ + lr) * INTER;    // token row
  const float* w2r = w2 + ((size_t)e * DIM + i) * INTER;     // weight row i
  const int koff = 2 * half;

  v8f acc = {};
  for (int k0 = 0; k0 < INTER; k0 += 8) {
    v2f a0 = *(const v2f*)(hr  + k0 + koff);
    v2f b0 = *(const v2f*)(w2r + k0 + koff);
    acc = wmma_f32(a0, b0, acc);
    v2f a1 = *(const v2f*)(hr  + k0 + 4 + koff);
    v2f b1 = *(const v2f*)(w2r + k0 + 4 + koff);
    acc = wmma_f32(a1, b1, acc);
  }

#pragma unroll
  for (int j = 0; j < 8; ++j) {
    const int t = j + 8 * half;
    const float v = acc[j];
    if (idx[t * TOPK + 0] == e) out[((size_t)t * TOPK + 0) * DIM + i] = v;
    if (idx[t * TOPK + 1] == e) out[((size_t)t * TOPK + 1) * DIM + i] = v;
  }
}

extern "C" void kernel_launch(void* const* d_in, const int* in_sizes, int n_in,
                              void* d_out, int out_size, void* d_ws,
                              size_t ws_size, hipStream_t stream) {
  const float* x  = (const float*)d_in[0];
  const int* idx  = (const int*)d_in[1];
  const float* w1 = (const float*)d_in[2];
  const float* w2 = (const float*)d_in[3];
  const float* w3 = (const float*)d_in[4];
  float* out = (float*)d_out;
  float* h   = (float*)d_ws;  // needs E*SEQ*INTER*4 = 3.67 MB

  dim3 g1(INTER / 128, NUM_EXPERTS);  // 56 x 8 blocks, 8 waves each
  moe_h_kernel<<<g1, 256, 0, stream>>>(x, w1, w3, h);

  dim3 g2(DIM / 128, NUM_EXPERTS);    // 16 x 8 blocks
  moe_out_kernel<<<g2, 256, 0, stream>>>(h, w2, idx, out);
}